// StateSpaceSimulator_68719477078
// MI455X (gfx1250) — compile-verified
//
#include <hip/hip_runtime.h>
#include <math.h>

// Reference dims
#define LL 1000
#define NN 2048
#define NX 16
#define NU 8
#define NF 128

// LDS strides (floats), padded to dodge bank conflicts (64 banks x 4B)
#define HS 132
#define XS 20

typedef __attribute__((ext_vector_type(2))) float v2f;
typedef __attribute__((ext_vector_type(8))) float v8f;

__device__ __forceinline__ float dev_tanh(float x) {
#if __has_builtin(__builtin_amdgcn_tanhf)
    return __builtin_amdgcn_tanhf(x);           // native v_tanh_f32 on gfx1250
#elif __has_builtin(__builtin_amdgcn_tanh_f32)
    return __builtin_amdgcn_tanh_f32(x);
#else
    return tanhf(x);
#endif
}

// One wave (32 lanes) owns a 16-row batch tile for the whole 1000-step scan.
// x state lives in VGPRs in WMMA C/D layout; weights live in VGPRs in WMMA
// B layout; D->A transposes between the two GEMMs go through per-wave LDS.
__global__ __launch_bounds__(32, 1)
void ssm_scan_kernel(const float* __restrict__ x0,
                     const float* __restrict__ u,
                     const float* __restrict__ W1,
                     const float* __restrict__ b1,
                     const float* __restrict__ W2,
                     const float* __restrict__ b2,
                     float* __restrict__ out)
{
    __shared__ float hlds[16 * HS];   // 16 x 128 hidden, padded
    __shared__ float xlds[16 * XS];   // 16 x 16 state, padded

    const int ln = threadIdx.x & 15;      // column/row-in-half index
    const int g  = threadIdx.x >> 4;      // lane half (0 or 1)
    const int m0 = blockIdx.x * 16;       // batch-row base of this tile

    // ---- Load weights into WMMA B-operand layout (loop invariant) ----
    // B 4x16 chunk: lanes 0-15 hold K=+0,+1 in v.x,v.y; lanes 16-31 hold K=+2,+3
    v2f w1b[6][8];
#pragma unroll
    for (int k = 0; k < 6; ++k) {
        const int r = 4 * k + 2 * g;
#pragma unroll
        for (int t = 0; t < 8; ++t) {
            v2f w;
            w.x = W1[(r + 0) * NF + 16 * t + ln];
            w.y = W1[(r + 1) * NF + 16 * t + ln];
            w1b[k][t] = w;
        }
    }
    v2f w2b[32];
#pragma unroll
    for (int k = 0; k < 32; ++k) {
        const int r = 4 * k + 2 * g;
        v2f w;
        w.x = W2[(r + 0) * NX + ln];
        w.y = W2[(r + 1) * NX + ln];
        w2b[k] = w;
    }
    float b1v[8];
#pragma unroll
    for (int t = 0; t < 8; ++t) b1v[t] = b1[16 * t + ln];
    const float b2v = b2[ln];

    // ---- Load initial state in C/D layout: xr[v] = x[m0 + v + 8g][ln] ----
    v8f xr;
#pragma unroll
    for (int v = 0; v < 8; ++v) xr[v] = x0[(size_t)(m0 + v + 8 * g) * NX + ln];

    const float* urow = u + (size_t)(m0 + ln) * NU;   // += NN*NU per step
    float*       orow = out + (size_t)m0 * NX;        // += NN*NX per step

#pragma unroll 1
    for (int l = 0; l < LL; ++l) {
        // Emit pre-update state (coalesced: each half-wave writes 64B rows)
#pragma unroll
        for (int v = 0; v < 8; ++v)
            orow[(v + 8 * g) * NX + ln] = xr[v];

        // Stage x (D layout) into LDS for the D->A transpose
#pragma unroll
        for (int v = 0; v < 8; ++v)
            xlds[(v + 8 * g) * XS + ln] = xr[v];

        // u columns for A-chunks 4 and 5 (xu cols 16..23)
        const v2f a4 = *(const v2f*)(urow + 2 * g);
        const v2f a5 = *(const v2f*)(urow + 4 + 2 * g);

        __syncthreads();  // order LDS stores before cross-lane loads

        // A operand chunks for GEMM1: lane reads row ln, cols 4k+2g..4k+2g+1
        v2f a[6];
#pragma unroll
        for (int k = 0; k < 4; ++k)
            a[k] = *(const v2f*)(&xlds[ln * XS + 4 * k + 2 * g]);
        a[4] = a4;
        a[5] = a5;

        // ---- GEMM1: h = xu @ W1 + b1  (8 independent N-tiles, K chains of 6)
        v8f hacc[8];
#pragma unroll
        for (int t = 0; t < 8; ++t) {
            v8f c;
#pragma unroll
            for (int v = 0; v < 8; ++v) c[v] = b1v[t];   // bias as C-init
#pragma unroll
            for (int k = 0; k < 6; ++k)
                c = __builtin_amdgcn_wmma_f32_16x16x4_f32(
                        false, a[k], false, w1b[k][t], (short)0, c,
                        false, false);
            hacc[t] = c;
        }

        // ---- tanh, then stage h (D layout) into LDS for D->A transpose ----
#pragma unroll
        for (int t = 0; t < 8; ++t) {
#pragma unroll
            for (int v = 0; v < 8; ++v)
                hlds[(v + 8 * g) * HS + 16 * t + ln] = dev_tanh(hacc[t][v]);
        }

        __syncthreads();  // order LDS stores before cross-lane loads

        // ---- GEMM2: dx = h @ W2 + b2, K=128 split over 4 partial accs ----
        v8f dacc[4];
#pragma unroll
        for (int v = 0; v < 8; ++v) dacc[0][v] = b2v;    // bias as C-init
#pragma unroll
        for (int p = 1; p < 4; ++p)
#pragma unroll
            for (int v = 0; v < 8; ++v) dacc[p][v] = 0.0f;

#pragma unroll
        for (int k = 0; k < 32; ++k) {
            const v2f a2 = *(const v2f*)(&hlds[ln * HS + 4 * k + 2 * g]);
            dacc[k & 3] = __builtin_amdgcn_wmma_f32_16x16x4_f32(
                              false, a2, false, w2b[k], (short)0, dacc[k & 3],
                              false, false);
        }

        const v8f dx = (dacc[0] + dacc[1]) + (dacc[2] + dacc[3]);
        xr += dx;                         // x_{l+1} = x_l + dx (C/D layout)

        __syncthreads();  // WAR guard: next step's LDS stores vs this step's loads

        urow += (size_t)NN * NU;
        orow += (size_t)NN * NX;
    }
}

extern "C" void kernel_launch(void* const* d_in, const int* in_sizes, int n_in,
                              void* d_out, int out_size, void* d_ws, size_t ws_size,
                              hipStream_t stream) {
    (void)in_sizes; (void)n_in; (void)out_size; (void)d_ws; (void)ws_size;
    const float* x0 = (const float*)d_in[0];
    const float* u  = (const float*)d_in[1];
    const float* W1 = (const float*)d_in[2];
    const float* b1 = (const float*)d_in[3];
    const float* W2 = (const float*)d_in[4];
    const float* b2 = (const float*)d_in[5];
    float* out = (float*)d_out;

    // 128 independent single-wave workgroups, one 16-row batch tile each.
    ssm_scan_kernel<<<NN / 16, 32, 0, stream>>>(x0, u, W1, b1, W2, b2, out);
}